// Custom_criterion1_84945863180460
// MI455X (gfx1250) — compile-verified
//
#include <hip/hip_runtime.h>
#include <hip/hip_bf16.h>

typedef float v2f __attribute__((ext_vector_type(2)));
typedef float v8f __attribute__((ext_vector_type(8)));

#define IMG_H   512
#define IMG_W   512
#define OUT_HW  497            // 512 - 16 + 1
#define N_IMG   96             // 32 batch * 3 channels
#define WIN     16
#define C1F     6.5025f        // (0.01*255)^2
#define C2F     58.5225f       // (0.03*255)^2

__global__ void ssim_init_acc(double* acc) {
    acc[0] = 0.0;   // mse sum
    acc[1] = 0.0;   // ssim sum
}

// ---------------------------------------------------------------------------
// MSE partial sums: grid-stride over float4, double accumulation.
// ---------------------------------------------------------------------------
__global__ void ssim_mse_partial(const float* __restrict__ x,
                                 const float* __restrict__ y,
                                 double* __restrict__ acc, int n4) {
    const float4* __restrict__ x4 = (const float4*)x;
    const float4* __restrict__ y4 = (const float4*)y;
    double sum = 0.0;
    for (int i = blockIdx.x * blockDim.x + threadIdx.x; i < n4;
         i += gridDim.x * blockDim.x) {
        float4 a = x4[i];
        float4 b = y4[i];
        float d0 = a.x - b.x, d1 = a.y - b.y, d2 = a.z - b.z, d3 = a.w - b.w;
        sum += (double)(d0 * d0 + d1 * d1 + d2 * d2 + d3 * d3);
    }
    // wave32 reduction
    for (int off = 16; off > 0; off >>= 1)
        sum += __shfl_xor(sum, off, 32);
    __shared__ double red[8];
    int w = threadIdx.x >> 5, l = threadIdx.x & 31;
    if (l == 0) red[w] = sum;
    __syncthreads();
    if (threadIdx.x == 0) {
        double t = 0.0;
        int nw = (blockDim.x + 31) >> 5;
        for (int i = 0; i < nw; ++i) t += red[i];
        atomicAdd(&acc[0], t);
    }
}

// ---------------------------------------------------------------------------
// Fused SSIM kernel: one wave32 per 16x16 output tile of one channel-image.
// Separable 16-tap Gaussian conv of {x, y, x^2, y^2, xy} via banded-matrix
// matmuls on V_WMMA_F32_16X16X4_F32 (K=32 chains of 8 wmma ops).
// The banded Gaussian matrix W[k][n] = g[k-n] is materialized in LDS once,
// so all fragment fetches are unconditional ds_loads (no EXEC-masked loads).
//   grid = (32, 32, 96), block = 32 threads (one wave).
// ---------------------------------------------------------------------------
__global__ void __launch_bounds__(32)
ssim_tile_wmma(const float* __restrict__ imgX,
               const float* __restrict__ imgY,
               double* __restrict__ acc) {
    __shared__ float sX[32][33];        // input patch, scaled to [0,255]
    __shared__ float sY[32][33];
    __shared__ float sH[5][32][17];     // horizontal conv results per field
    __shared__ float sG[WIN];           // 1D gaussian, normalized
    __shared__ float sW[32][17];        // banded matrix: sW[k][n] = g[k-n]

    const int lane = threadIdx.x & 31;
    const int tileC = blockIdx.x;       // 0..31
    const int tileR = blockIdx.y;       // 0..31
    const int img   = blockIdx.z;       // 0..95
    const int R0 = tileR * 16;
    const int C0 = tileC * 16;

    const float* __restrict__ px = imgX + (size_t)img * IMG_H * IMG_W;
    const float* __restrict__ py = imgY + (size_t)img * IMG_H * IMG_W;

    // --- 1D gaussian window (sigma=1.5), normalized to sum 1 ---
    if (lane < WIN) {
        float c = (float)lane - 7.5f;
        sG[lane] = __expf(-(c * c) / 4.5f);
    }
    __syncthreads();
    float gtot = 0.0f;
    #pragma unroll
    for (int i = 0; i < WIN; ++i) gtot += sG[i];
    __syncthreads();
    if (lane < WIN) sG[lane] = sG[lane] / gtot;
    __syncthreads();

    // --- materialize banded gaussian matrix (branch-free: load + select) ---
    #pragma unroll
    for (int it = 0; it < 16; ++it) {
        int idx = it * 32 + lane;       // 0..511
        int k = idx >> 4, n = idx & 15;
        int d = k - n;
        float gv = sG[d & 15];          // unconditional LDS read
        sW[k][n] = (d >= 0 && d < WIN) ? gv : 0.0f;
    }

    // --- load 31x31 input patch (row 31 zero-padded; col 31 loaded, zero weight; OOB -> 0) ---
    for (int r = 0; r < 32; ++r) {
        int gr = R0 + r, gc = C0 + lane;
        bool ok = (r < 31) && (gr < IMG_H) && (gc < IMG_W);
        float xv = 0.0f, yv = 0.0f;
        if (ok) {
            size_t o = (size_t)gr * IMG_W + gc;
            xv = px[o] * 255.0f;
            yv = py[o] * 255.0f;
        }
        sX[r][lane] = xv;
        sY[r][lane] = yv;
    }
    __syncthreads();

    const int hi  = lane >> 4;      // 0 or 1 (half-wave)
    const int m   = lane & 15;      // row / col index within fragment
    const int hi8 = hi * 8;

    // ================= Stage 1: horizontal conv  H = P * W ==================
    // H[j][c] = sum_k P[j][k] * g[k-c], j = patch row (two 16-row chunks).
    #pragma unroll
    for (int chunk = 0; chunk < 2; ++chunk) {
        v8f a0 = {0,0,0,0,0,0,0,0};  // x
        v8f a1 = {0,0,0,0,0,0,0,0};  // y
        v8f a2 = {0,0,0,0,0,0,0,0};  // x*x
        v8f a3 = {0,0,0,0,0,0,0,0};  // y*y
        v8f a4 = {0,0,0,0,0,0,0,0};  // x*y
        #pragma unroll
        for (int s = 0; s < 8; ++s) {
            const int kb = 4 * s;
            const int kA = kb + 2 * hi;          // A: lanes 16-31 carry K+2,K+3
            const int row = chunk * 16 + m;
            float x0 = sX[row][kA],  x1 = sX[row][kA + 1];
            float y0 = sY[row][kA],  y1 = sY[row][kA + 1];
            v2f aX  = { x0, x1 };
            v2f aY  = { y0, y1 };
            v2f aXX = { x0 * x0, x1 * x1 };
            v2f aYY = { y0 * y0, y1 * y1 };
            v2f aXY = { x0 * y0, x1 * y1 };
            // B fragment: banded gaussian W[k][n] = g[k-n]
            v2f b = { sW[kA][m], sW[kA + 1][m] };
            a0 = __builtin_amdgcn_wmma_f32_16x16x4_f32(false, aX,  false, b, (short)0, a0, false, false);
            a1 = __builtin_amdgcn_wmma_f32_16x16x4_f32(false, aY,  false, b, (short)0, a1, false, false);
            a2 = __builtin_amdgcn_wmma_f32_16x16x4_f32(false, aXX, false, b, (short)0, a2, false, false);
            a3 = __builtin_amdgcn_wmma_f32_16x16x4_f32(false, aYY, false, b, (short)0, a3, false, false);
            a4 = __builtin_amdgcn_wmma_f32_16x16x4_f32(false, aXY, false, b, (short)0, a4, false, false);
        }
        // store H fragments: D layout: vgpr v -> M = v (lanes 0-15) / v+8
        #pragma unroll
        for (int v = 0; v < 8; ++v) {
            int j = chunk * 16 + v + hi8;
            sH[0][j][m] = a0[v];
            sH[1][j][m] = a1[v];
            sH[2][j][m] = a2[v];
            sH[3][j][m] = a3[v];
            sH[4][j][m] = a4[v];
        }
        __syncthreads();
    }

    // ================= Stage 2: vertical conv  D = W2 * H ===================
    // out[r][c] = sum_j g[j-r] * H[j][c];  W2[r][j] = g[j-r] = sW[j][r]
    v8f mMu1 = {0,0,0,0,0,0,0,0};
    v8f mMu2 = {0,0,0,0,0,0,0,0};
    v8f mXX  = {0,0,0,0,0,0,0,0};
    v8f mYY  = {0,0,0,0,0,0,0,0};
    v8f mXY  = {0,0,0,0,0,0,0,0};
    #pragma unroll
    for (int s = 0; s < 8; ++s) {
        const int kb = 4 * s;
        const int kA = kb + 2 * hi;
        // A fragment: banded gaussian transpose, same table
        v2f a = { sW[kA][m], sW[kA + 1][m] };
        // B fragments from H (K = j, N = c)
        v2f b0 = { sH[0][kA][m], sH[0][kA + 1][m] };
        v2f b1 = { sH[1][kA][m], sH[1][kA + 1][m] };
        v2f b2 = { sH[2][kA][m], sH[2][kA + 1][m] };
        v2f b3 = { sH[3][kA][m], sH[3][kA + 1][m] };
        v2f b4 = { sH[4][kA][m], sH[4][kA + 1][m] };
        mMu1 = __builtin_amdgcn_wmma_f32_16x16x4_f32(false, a, false, b0, (short)0, mMu1, false, false);
        mMu2 = __builtin_amdgcn_wmma_f32_16x16x4_f32(false, a, false, b1, (short)0, mMu2, false, false);
        mXX  = __builtin_amdgcn_wmma_f32_16x16x4_f32(false, a, false, b2, (short)0, mXX,  false, false);
        mYY  = __builtin_amdgcn_wmma_f32_16x16x4_f32(false, a, false, b3, (short)0, mYY,  false, false);
        mXY  = __builtin_amdgcn_wmma_f32_16x16x4_f32(false, a, false, b4, (short)0, mXY,  false, false);
    }

    // --- per-element SSIM + masked accumulation ---
    float lsum = 0.0f;
    #pragma unroll
    for (int v = 0; v < 8; ++v) {
        float mu1 = mMu1[v], mu2 = mMu2[v];
        float mu11 = mu1 * mu1, mu22 = mu2 * mu2, mu12 = mu1 * mu2;
        float s11 = mXX[v] - mu11;
        float s22 = mYY[v] - mu22;
        float s12 = mXY[v] - mu12;
        float num = (2.0f * mu12 + C1F) * (2.0f * s12 + C2F);
        float den = (mu11 + mu22 + C1F) * (s11 + s22 + C2F);
        float sv = num / den;
        int orow = R0 + v + hi8;   // D layout: M = v + 8*hi
        int ocol = C0 + m;         // N = lane & 15
        if (orow < OUT_HW && ocol < OUT_HW) lsum += sv;
    }
    for (int off = 16; off > 0; off >>= 1)
        lsum += __shfl_xor(lsum, off, 32);
    if (lane == 0) atomicAdd(&acc[1], (double)lsum);
}

__global__ void ssim_finalize(const double* __restrict__ acc,
                              float* __restrict__ out) {
    const double NTOT = 32.0 * 3.0 * 512.0 * 512.0;
    const double MTOT = (double)N_IMG * OUT_HW * OUT_HW;
    double mse  = acc[0] / NTOT;
    double ssim = acc[1] / MTOT;
    out[0] = (float)(0.7 * mse + 0.3 * (1.0 - ssim));
}

extern "C" void kernel_launch(void* const* d_in, const int* in_sizes, int n_in,
                              void* d_out, int out_size, void* d_ws, size_t ws_size,
                              hipStream_t stream) {
    const float* x = (const float*)d_in[0];   // output
    const float* y = (const float*)d_in[1];   // target
    float* out = (float*)d_out;
    double* acc = (double*)d_ws;

    ssim_init_acc<<<1, 1, 0, stream>>>(acc);

    int n4 = (32 * 3 * 512 * 512) / 4;
    ssim_mse_partial<<<4096, 256, 0, stream>>>(x, y, acc, n4);

    dim3 grid(32, 32, N_IMG);
    ssim_tile_wmma<<<grid, 32, 0, stream>>>(x, y, acc);

    ssim_finalize<<<1, 1, 0, stream>>>(acc, out);
}